// InstructedAttentionPositionScores_72189810311267
// MI455X (gfx1250) — compile-verified
//
#include <hip/hip_runtime.h>
#include <hip/hip_bf16.h>

typedef __attribute__((ext_vector_type(2))) float v2f;
typedef __attribute__((ext_vector_type(4))) float v4f;
typedef __attribute__((ext_vector_type(8))) float v8f;

// ---------------- constants (from reference setup) ----------------
#define EMB    64
#define HEADS  8
#define TOK    10
#define DIM_I  100
#define DIM_Q  4708
#define DIM_H  24
#define DIM_W  24
#define DIM_D  8
// workspace table offsets (floats)
#define OFF_INTRA 0        // [8][10][10]
#define OFF_INTER 800      // [8][10][10]
#define OFF_CIC   1600     // [8][10]
#define OFF_CCI   1680     // [8][10]
#define OFF_RS    1760     // [8][24][24]
#define OFF_CS    6368     // [8][24][24]
#define OFF_DS    10976    // [8][8][8]

// =====================================================================
// Phase 1: build the small score tables with V_WMMA_F32_16X16X4_F32.
// Each wave computes a 16(heads, padded)x16(p) tile of out[h,p] =
// sum_c W[h,c] * E[row(p), c], accumulating K=64 in 16 chained WMMAs.
//   A (16x4 f32):  lane<16 -> {K0,K1}, lane>=16 -> {K2,K3}   (per ISA)
//   B (4x16 f32):  VGPR v, lane-half sel -> K = v + 2*half     (C/D-style)
//   D (16x16 f32): lane L, vgpr v -> row v + 8*(L>>4), col L&15
// Heads 0..7 live in lanes 0..15, vgprs 0..7. All loads are branchless:
// padded A rows are clamped to a valid row and zeroed via v_cndmask so
// EXEC stays all-ones through the whole WMMA chain.
// =====================================================================
__global__ __launch_bounds__(32)
void iaps_phase1_tables(const float* __restrict__ enc_intra,
                        const float* __restrict__ enc_inter,
                        const float* __restrict__ enc_cic,
                        const float* __restrict__ enc_cci,
                        const float* __restrict__ enc_h,
                        const float* __restrict__ enc_w,
                        const float* __restrict__ enc_d,
                        const float* __restrict__ w_intra,
                        const float* __restrict__ w_inter,
                        const float* __restrict__ w_cic,
                        const float* __restrict__ w_cci,
                        const float* __restrict__ w_h,
                        const float* __restrict__ w_w,
                        const float* __restrict__ w_d,
                        float* __restrict__ ws)
{
    const int tile = blockIdx.x;   // 92 tiles total
    int tbase, P, dim, cap;
    const float *W, *E;
    float* dst;
    // tile ranges: intra[0,7) inter[7,14) cic[14,15) cci[15,16)
    //              rs[16,52) cs[52,88) ds[88,92)
    if (tile < 7)       { tbase = 0;  W = w_intra; E = enc_intra; dst = ws + OFF_INTRA; P = 100; dim = 0;  cap = 0;  }
    else if (tile < 14) { tbase = 7;  W = w_inter; E = enc_inter; dst = ws + OFF_INTER; P = 100; dim = 0;  cap = 0;  }
    else if (tile < 15) { tbase = 14; W = w_cic;   E = enc_cic;   dst = ws + OFF_CIC;   P = 10;  dim = 0;  cap = 0;  }
    else if (tile < 16) { tbase = 15; W = w_cci;   E = enc_cci;   dst = ws + OFF_CCI;   P = 10;  dim = 0;  cap = 0;  }
    else if (tile < 52) { tbase = 16; W = w_h;     E = enc_h;     dst = ws + OFF_RS;    P = 576; dim = 24; cap = 64; }
    else if (tile < 88) { tbase = 52; W = w_w;     E = enc_w;     dst = ws + OFF_CS;    P = 576; dim = 24; cap = 64; }
    else                { tbase = 88; W = w_d;     E = enc_d;     dst = ws + OFF_DS;    P = 64;  dim = 8;  cap = 8;  }

    const int lane = threadIdx.x;       // 0..31
    const int half = lane >> 4;         // 0: K 0,1   1: K 2,3
    const int nm   = lane & 15;         // A row (head) / B col (p)
    const int p0   = (tile - tbase) * 16;

    int  p     = p0 + nm;
    bool valid = (p < P);
    if (!valid) p = P - 1;              // clamp gather; store masked below

    // Gather the embedding row for column p (relative-index modes fold the
    // clip(d + cap - 1, 0, 2*cap-2) lookup into the row pointer).
    const float* erow;
    if (dim == 0) {
        erow = E + p * EMB;
    } else {
        int n   = p / dim;
        int m   = p - n * dim;
        int idx = m - n + (cap - 1);
        int tl  = 2 * cap - 2;
        idx = idx < 0 ? 0 : (idx > tl ? tl : idx);
        erow = E + idx * EMB;
    }

    // A rows 8..15 are zero-padding: clamp the pointer into valid memory so
    // the load is unconditional, then zero the value with a select.
    const bool wlive = (nm < HEADS);
    const v2f* __restrict__ w2 = (const v2f*)(W + (nm & (HEADS - 1)) * EMB);
    const v2f* __restrict__ e2 = (const v2f*)erow;

    v8f c = {0.f, 0.f, 0.f, 0.f, 0.f, 0.f, 0.f, 0.f};
#pragma unroll
    for (int kb = 0; kb < 16; ++kb) {
        const int ci = kb * 2 + half;   // v2f index of this lane's K-pair
        v2f a = w2[ci];                 // global_load_b64, always in-bounds
        v2f b = e2[ci];                 // global_load_b64
        a.x = wlive ? a.x : 0.f;        // v_cndmask, no EXEC games
        a.y = wlive ? a.y : 0.f;
        c = __builtin_amdgcn_wmma_f32_16x16x4_f32(
                /*neg_a=*/false, a, /*neg_b=*/false, b,
                /*c_mod=*/(short)0, c, /*reuse_a=*/false, /*reuse_b=*/false);
    }

    if (half == 0 && valid) {
#pragma unroll
        for (int h = 0; h < HEADS; ++h)
            dst[h * P + p] = c[h];
    }
}

// =====================================================================
// Phase 2: stream 709 MB of output (store-bandwidth bound, ~30 us floor
// at 23.3 TB/s). One block per (h, q) row; 128-bit non-temporal stores.
// Content rows stage cs+ds (192 f) and the rs row (24 f) in LDS, so each
// element is one LDS read + broadcast + fma. Region split at k==100 is
// vector-aligned; each row base is 16B aligned (4708 % 4 == 0).
// =====================================================================
__global__ __launch_bounds__(256)
void iaps_phase2_fill(const float* __restrict__ ws, float* __restrict__ out)
{
    const float SCALE  = 0.125f;            // EMB^-0.5
    const float CSCALE = 0.125f / 3.0f;     // content also has /3

    const float* t_intra = ws + OFF_INTRA;
    const float* t_inter = ws + OFF_INTER;
    const float* t_cic   = ws + OFF_CIC;
    const float* t_cci   = ws + OFF_CCI;
    const float* t_rs    = ws + OFF_RS;
    const float* t_cs    = ws + OFF_CS;
    const float* t_ds    = ws + OFF_DS;

    const int row = blockIdx.x;             // 0 .. 8*4708-1
    const int h   = row / DIM_Q;
    const int q   = row - h * DIM_Q;
    v4f* __restrict__ orow4 = (v4f*)(out + (size_t)row * DIM_Q);
    const int NV4 = DIM_Q / 4;              // 1177 vec4 per row

    __shared__ float s192[DIM_W * DIM_D];   // cs_row[j2] + ds_row[j3]
    __shared__ float srs[DIM_H];            // rs_row[j1]

    if (q >= DIM_I) {
        // ---- content row: q -> (i1, i2, i3) ----
        const int qq = q - DIM_I;
        const int i1 = qq / (DIM_W * DIM_D);
        const int i2 = (qq / DIM_D) % DIM_W;
        const int i3 = qq & (DIM_D - 1);
        const float* cs_row = t_cs + h * (DIM_W * DIM_W) + i2 * DIM_W;
        const float* ds_row = t_ds + h * (DIM_D * DIM_D) + i3 * DIM_D;
        const float* rs_row = t_rs + h * (DIM_H * DIM_H) + i1 * DIM_H;

        const int t = threadIdx.x;
        if (t < 192)       s192[t] = cs_row[t >> 3] + ds_row[t & 7];
        else if (t < 216)  srs[t - 192] = rs_row[t - 192];
        __syncthreads();

        const float* cci_h = t_cci + h * TOK;
        for (int i4 = threadIdx.x; i4 < NV4; i4 += 256) {
            v4f v;
            const int k = i4 * 4;
            if (i4 < DIM_I / 4) {           // bottom-left: cci broadcast
                v.x = cci_h[(k    ) % TOK] * SCALE;
                v.y = cci_h[(k + 1) % TOK] * SCALE;
                v.z = cci_h[(k + 2) % TOK] * SCALE;
                v.w = cci_h[(k + 3) % TOK] * SCALE;
            } else {                        // content block
                const int kk   = k - DIM_I;             // multiple of 4
                const int j1   = kk / (DIM_W * DIM_D);
                const int j23  = kk % (DIM_W * DIM_D);  // stays <189, +3 safe
                const float r  = srs[j1];
                v.x = (r + s192[j23    ]) * CSCALE;
                v.y = (r + s192[j23 + 1]) * CSCALE;
                v.z = (r + s192[j23 + 2]) * CSCALE;
                v.w = (r + s192[j23 + 3]) * CSCALE;
            }
            __builtin_nontemporal_store(v, &orow4[i4]);
        }
    } else {
        // ---- instruction row: intra/inter block then cic broadcast ----
        const int a  = q / TOK;
        const int ar = q - a * TOK;
        const float  cicv = t_cic[h * TOK + ar] * SCALE;
        const float* ti = t_intra + h * (TOK * TOK) + ar * TOK;
        const float* tn = t_inter + h * (TOK * TOK) + ar * TOK;

        for (int i4 = threadIdx.x; i4 < NV4; i4 += 256) {
            v4f v;
            if (i4 < DIM_I / 4) {
                const int k = i4 * 4;
                float tmp[4];
#pragma unroll
                for (int j = 0; j < 4; ++j) {
                    const int kj = k + j;
                    const int b  = kj / TOK;
                    const int br = kj - b * TOK;
                    tmp[j] = ((b == a) ? ti[br] : tn[br]) * SCALE;
                }
                v.x = tmp[0]; v.y = tmp[1]; v.z = tmp[2]; v.w = tmp[3];
            } else {
                v.x = cicv; v.y = cicv; v.z = cicv; v.w = cicv;
            }
            __builtin_nontemporal_store(v, &orow4[i4]);
        }
    }
}

extern "C" void kernel_launch(void* const* d_in, const int* in_sizes, int n_in,
                              void* d_out, int out_size, void* d_ws, size_t ws_size,
                              hipStream_t stream) {
    const float* enc_intra = (const float*)d_in[0];
    const float* enc_inter = (const float*)d_in[1];
    const float* enc_cic   = (const float*)d_in[2];
    const float* enc_cci   = (const float*)d_in[3];
    const float* enc_h     = (const float*)d_in[4];
    const float* enc_w     = (const float*)d_in[5];
    const float* enc_d     = (const float*)d_in[6];
    const float* w_intra   = (const float*)d_in[7];
    const float* w_inter   = (const float*)d_in[8];
    const float* w_cic     = (const float*)d_in[9];
    const float* w_cci     = (const float*)d_in[10];
    const float* w_h       = (const float*)d_in[11];
    const float* w_w       = (const float*)d_in[12];
    const float* w_d       = (const float*)d_in[13];
    float* ws  = (float*)d_ws;
    float* out = (float*)d_out;

    // Phase 1: 92 tiles, one wave32 each (16 chained f32 WMMAs per wave).
    iaps_phase1_tables<<<92, 32, 0, stream>>>(
        enc_intra, enc_inter, enc_cic, enc_cci, enc_h, enc_w, enc_d,
        w_intra, w_inter, w_cic, w_cci, w_h, w_w, w_d, ws);

    // Phase 2: one block per (h, q) output row; NT 128-bit streaming stores.
    iaps_phase2_fill<<<HEADS * DIM_Q, 256, 0, stream>>>(ws, out);
}